// HistogramRangeTracker_53042846105724
// MI455X (gfx1250) — compile-verified
//
#include <hip/hip_runtime.h>

#define NBINS 256

typedef __attribute__((ext_vector_type(2))) float v2f;
typedef __attribute__((ext_vector_type(8))) float v8f;

// ---- monotonic float <-> uint key mapping (total order matches float order) ----
__device__ __forceinline__ unsigned fkey(float f) {
    unsigned u = __float_as_uint(f);
    return (u & 0x80000000u) ? ~u : (u | 0x80000000u);
}
__device__ __forceinline__ float kinv(unsigned k) {
    return __uint_as_float((k & 0x80000000u) ? (k ^ 0x80000000u) : ~k);
}

// ws layout (uint): [0]=min key, [1]=max key, [2..257]=histogram bins
__global__ void hrt_init(unsigned* __restrict__ mm, unsigned* __restrict__ gh) {
    int t = threadIdx.x;
    if (t < NBINS) gh[t] = 0u;
    if (t == 0) { mm[0] = 0xFFFFFFFFu; mm[1] = 0u; }
}

// ---- pass 1: global min/max, forward stream (leaves tensor tail hot in L2) ----
__global__ void hrt_minmax(const float4* __restrict__ x, size_t n4,
                           unsigned* __restrict__ mm) {
    float lo =  3.402823466e38f;
    float hi = -3.402823466e38f;
    size_t stride = (size_t)gridDim.x * blockDim.x;
    for (size_t i = (size_t)blockIdx.x * blockDim.x + threadIdx.x; i < n4; i += stride) {
        float4 v = x[i];
        lo = fminf(lo, fminf(fminf(v.x, v.y), fminf(v.z, v.w)));
        hi = fmaxf(hi, fmaxf(fmaxf(v.x, v.y), fmaxf(v.z, v.w)));
    }
    // wave32 butterfly reduction
    for (int o = 16; o > 0; o >>= 1) {
        lo = fminf(lo, __shfl_xor(lo, o, 32));
        hi = fmaxf(hi, __shfl_xor(hi, o, 32));
    }
    __shared__ float slo[8], shi[8];
    int wave = threadIdx.x >> 5;
    int lane = threadIdx.x & 31;
    if (lane == 0) { slo[wave] = lo; shi[wave] = hi; }
    __syncthreads();
    if (threadIdx.x == 0) {
        int nw = blockDim.x >> 5;
        for (int w = 1; w < nw; ++w) { lo = fminf(lo, slo[w]); hi = fmaxf(hi, shi[w]); }
        atomicMin(&mm[0], fkey(lo));
        atomicMax(&mm[1], fkey(hi));
    }
}

__device__ __forceinline__ int hrt_bin(float v, float tmin, float scale) {
    int b = (int)((v - tmin) * scale);       // v_cvt_i32_f32 truncates, matches astype(int32)
    return b < 0 ? 0 : (b > (NBINS - 1) ? (NBINS - 1) : b);
}

// ---- pass 2: histogram, REVERSE stream to hit the ~192MB of L2 left by pass 1 ----
__global__ void hrt_hist(const float4* __restrict__ x, size_t n4,
                         const unsigned* __restrict__ mm, unsigned* __restrict__ gh) {
    __shared__ unsigned lh[8 * NBINS];       // per-wave privatized histograms (8 waves/block)
    for (int i = threadIdx.x; i < 8 * NBINS; i += blockDim.x) lh[i] = 0u;
    float tmin = kinv(mm[0]);
    float tmax = kinv(mm[1]);
    float scale = 256.0f / (tmax - tmin);
    __syncthreads();

    unsigned* h = &lh[(threadIdx.x >> 5) * NBINS];
    size_t stride = (size_t)gridDim.x * blockDim.x;
    for (size_t i = (size_t)blockIdx.x * blockDim.x + threadIdx.x; i < n4; i += stride) {
        float4 v = x[n4 - 1 - i];            // reversed: start where L2 is warm
        atomicAdd(&h[hrt_bin(v.x, tmin, scale)], 1u);
        atomicAdd(&h[hrt_bin(v.y, tmin, scale)], 1u);
        atomicAdd(&h[hrt_bin(v.z, tmin, scale)], 1u);
        atomicAdd(&h[hrt_bin(v.w, tmin, scale)], 1u);
    }
    __syncthreads();
    for (int b = threadIdx.x; b < NBINS; b += blockDim.x) {
        unsigned s = 0;
        for (int w = 0; w < 8; ++w) s += lh[w * NBINS + b];
        if (s) atomicAdd(&gh[b], s);
    }
}

// ---- pass 3: single wave. 256-element prefix sum via WMMA triangular matmuls:
//      M = hist viewed as 16x16;  cum = M*U + (Lstrict*M)*Ones
//      (3 GEMMs of 16x16x16 = 12 x v_wmma_f32_16x16x4_f32)
__global__ void hrt_final(const unsigned* __restrict__ mm, const unsigned* __restrict__ gh,
                          float* __restrict__ out) {
    __shared__ float sm[NBINS];
    const int lane = threadIdx.x;      // 0..31, full wave => EXEC all ones for WMMA
    const int hi   = lane >> 4;        // half-wave selector
    const int r    = lane & 15;        // A-layout row / B,C,D-layout column

    // P = M x U   (U[k][n] = k <= n : intra-row inclusive prefix)
    v8f p = {};
    for (int t = 0; t < 4; ++t) {
        int k0 = 4 * t + (hi ? 2 : 0);
        int k1 = 4 * t + (hi ? 3 : 1);
        v2f a, b;
        a.x = (float)gh[r * 16 + k0];
        a.y = (float)gh[r * 16 + k1];
        b.x = (k0 <= r) ? 1.0f : 0.0f;
        b.y = (k1 <= r) ? 1.0f : 0.0f;
        p = __builtin_amdgcn_wmma_f32_16x16x4_f32(false, a, false, b, (short)0, p, false, false);
    }
    // Q = Lstrict x M   (Q[r][c] = sum of rows above r)
    v8f q = {};
    for (int t = 0; t < 4; ++t) {
        int k0 = 4 * t + (hi ? 2 : 0);
        int k1 = 4 * t + (hi ? 3 : 1);
        v2f a, b;
        a.x = (k0 < r) ? 1.0f : 0.0f;
        a.y = (k1 < r) ? 1.0f : 0.0f;
        b.x = (float)gh[k0 * 16 + r];
        b.y = (float)gh[k1 * 16 + r];
        q = __builtin_amdgcn_wmma_f32_16x16x4_f32(false, a, false, b, (short)0, q, false, false);
    }
    // Q is in C/D layout; round-trip through LDS to refeed it in A layout
    for (int i = 0; i < 8; ++i) sm[(i + 8 * hi) * 16 + r] = q[i];
    __syncthreads();
    // cum = Q x Ones + P  (row-broadcast of Q's row sums, accumulated onto P)
    v8f c = p;
    for (int t = 0; t < 4; ++t) {
        int k0 = 4 * t + (hi ? 2 : 0);
        int k1 = 4 * t + (hi ? 3 : 1);
        v2f a, b;
        a.x = sm[r * 16 + k0];
        a.y = sm[r * 16 + k1];
        b.x = 1.0f;
        b.y = 1.0f;
        c = __builtin_amdgcn_wmma_f32_16x16x4_f32(false, a, false, b, (short)0, c, false, false);
    }
    __syncthreads();
    for (int i = 0; i < 8; ++i) sm[(i + 8 * hi) * 16 + r] = c[i];
    __syncthreads();

    if (lane == 0) {
        float total = sm[NBINS - 1];
        float tl = total * 0.005f;           // total*(1-0.99)/2
        float tu = total * 0.995f;           // total*(1+0.99)/2
        int li = 0, ui = 0;
        bool fl = false, fu = false;
        for (int i = 0; i < NBINS; ++i) {
            float cv = sm[i];
            if (!fl && cv > tl) { li = i; fl = true; }
            if (!fu && cv > tu) { ui = i; fu = true; }
        }
        float tmin = kinv(mm[0]);
        float tmax = kinv(mm[1]);
        float w = tmax - tmin;
        out[0] = tmin + w * ((float)li * (1.0f / 256.0f));   // linspace edge[li]
        out[1] = tmin + w * ((float)ui * (1.0f / 256.0f));   // linspace edge[ui]
    }
}

extern "C" void kernel_launch(void* const* d_in, const int* in_sizes, int n_in,
                              void* d_out, int out_size, void* d_ws, size_t ws_size,
                              hipStream_t stream) {
    const float* x = (const float*)d_in[0];
    size_t n  = (size_t)in_sizes[0];
    size_t n4 = n >> 2;                      // shape is 2^26, divisible by 4

    unsigned* ws = (unsigned*)d_ws;
    unsigned* mm = ws;                       // [0]=min key, [1]=max key
    unsigned* gh = ws + 2;                   // [2..257] histogram

    float* out = (float*)d_out;

    hrt_init<<<dim3(1), dim3(256), 0, stream>>>(mm, gh);
    hrt_minmax<<<dim3(1024), dim3(256), 0, stream>>>((const float4*)x, n4, mm);
    hrt_hist<<<dim3(1024), dim3(256), 0, stream>>>((const float4*)x, n4, mm, gh);
    hrt_final<<<dim3(1), dim3(32), 0, stream>>>(mm, gh, out);
}